// StyleTTS2ForONNX_77438260347180
// MI455X (gfx1250) — compile-verified
//
#include <hip/hip_runtime.h>
#include <hip/hip_bf16.h>
#include <stdint.h>

#define SEQ_LEN    2048
#define D_DIM      640
#define C_DIM      512
#define N_CH       (D_DIM + C_DIM)     // 1152
#define FRAMES     34816               // SEQ_LEN * 17
#define FR_PER_BLK 1024                // 256 threads * 4 frames
#define CH_PER_BLK 32

// Native clang vector type: accepted by __builtin_nontemporal_store and
// lowers to b128 global memory ops.
typedef __attribute__((ext_vector_type(4))) float vfloat4;

// ---------------------------------------------------------------------------
// Kernel 1: build frame -> token map in scratch.
//   map[f] = s  if start[s] <= f < end[s],  -1 for tail frames.
// Single workgroup; trivial vs. the 160 MB streaming kernel.
// ---------------------------------------------------------------------------
__global__ __launch_bounds__(256)
void build_map_kernel(const int* __restrict__ pred_dur, int* __restrict__ map)
{
    __shared__ int partial[256];
    __shared__ int scanbase[256];
    const int tid = threadIdx.x;

    // Phase 0: default the whole map to "no token".
    for (int i = tid; i < FRAMES; i += 256) map[i] = -1;

    // Phase 1: per-thread local durations (8 tokens each) + local sum.
    int dur[8];
    int sum = 0;
#pragma unroll
    for (int k = 0; k < 8; ++k) {
        int v = pred_dur[tid * 8 + k];
        v = (v < 1) ? 1 : v;           // clamp(min=1)
        dur[k] = v;
        sum += v;
    }
    partial[tid] = sum;
    __syncthreads();

    // Phase 2: exclusive scan of the 256 partials (tiny; serial on one lane).
    if (tid == 0) {
        int acc = 0;
        for (int i = 0; i < 256; ++i) { scanbase[i] = acc; acc += partial[i]; }
    }
    __syncthreads();

    // Phase 3: scatter token ids into the map.
    int start = scanbase[tid];
#pragma unroll
    for (int k = 0; k < 8; ++k) {
        const int s = tid * 8 + k;
        for (int j = 0; j < dur[k]; ++j) map[start + j] = s;
        start += dur[k];
    }
}

// ---------------------------------------------------------------------------
// Kernel 2: streaming gather-expand.
//   out[ch, f] = ch < 640 ? d[s(f)*640 + ch] : t_en[(ch-640)*2048 + s(f)]
// grid = (FRAMES/1024, N_CH/32), block = 256, 4 frames per thread.
// Map tile staged into LDS via the CDNA5 async-to-LDS path (ASYNCcnt);
// output written with non-temporal b128 stores so the 160 MB stream does not
// evict the L2-resident gather inputs.
// ---------------------------------------------------------------------------
__global__ __launch_bounds__(256)
void expand_kernel(const float* __restrict__ d,
                   const float* __restrict__ t_en,
                   const int*   __restrict__ map,
                   float*       __restrict__ out)
{
    __shared__ __align__(16) int smap[FR_PER_BLK];

    const int tid = threadIdx.x;
    const int f0  = blockIdx.x * FR_PER_BLK + tid * 4;   // 4 frames per thread

    // --- Async-stage this block's 1024 map entries into LDS (b128/lane). ---
    {
        uint32_t lds_addr = (uint32_t)(uintptr_t)(&smap[tid * 4]); // low 32b = LDS offset
        uint64_t gaddr    = (uint64_t)(uintptr_t)(map + f0);
        asm volatile("global_load_async_to_lds_b128 %0, %1, off"
                     :: "v"(lds_addr), "v"(gaddr) : "memory");
        asm volatile("s_wait_asynccnt 0x0" ::: "memory");
    }
    __syncthreads();

    const int4 s4 = *(const int4*)(&smap[tid * 4]);

    const int ch0 = blockIdx.y * CH_PER_BLK;
    float* outp = out + (size_t)ch0 * FRAMES + f0;

    const vfloat4 zero4 = {0.0f, 0.0f, 0.0f, 0.0f};

    if (ch0 < D_DIM) {
        // en part: gather rows of d (d[s*640 + ch]); vectorized 4ch x 4f with
        // an in-register 4x4 transpose: 4 b128 loads + 4 NT b128 stores per step.
        const float* __restrict__ dr0 = (s4.x >= 0) ? d + (size_t)s4.x * D_DIM : nullptr;
        const float* __restrict__ dr1 = (s4.y >= 0) ? d + (size_t)s4.y * D_DIM : nullptr;
        const float* __restrict__ dr2 = (s4.z >= 0) ? d + (size_t)s4.z * D_DIM : nullptr;
        const float* __restrict__ dr3 = (s4.w >= 0) ? d + (size_t)s4.w * D_DIM : nullptr;
#pragma unroll
        for (int c = 0; c < CH_PER_BLK; c += 4) {
            const int ch = ch0 + c;                      // ch % 4 == 0 -> 16B aligned
            const vfloat4 a0 = dr0 ? *(const vfloat4*)(dr0 + ch) : zero4; // frame f0+0
            const vfloat4 a1 = dr1 ? *(const vfloat4*)(dr1 + ch) : zero4; // frame f0+1
            const vfloat4 a2 = dr2 ? *(const vfloat4*)(dr2 + ch) : zero4; // frame f0+2
            const vfloat4 a3 = dr3 ? *(const vfloat4*)(dr3 + ch) : zero4; // frame f0+3

            vfloat4 v0 = {a0.x, a1.x, a2.x, a3.x};       // channel ch+0, frames f0..f0+3
            vfloat4 v1 = {a0.y, a1.y, a2.y, a3.y};
            vfloat4 v2 = {a0.z, a1.z, a2.z, a3.z};
            vfloat4 v3 = {a0.w, a1.w, a2.w, a3.w};

            __builtin_nontemporal_store(v0, (vfloat4*)(outp + (size_t)(c + 0) * FRAMES));
            __builtin_nontemporal_store(v1, (vfloat4*)(outp + (size_t)(c + 1) * FRAMES));
            __builtin_nontemporal_store(v2, (vfloat4*)(outp + (size_t)(c + 2) * FRAMES));
            __builtin_nontemporal_store(v3, (vfloat4*)(outp + (size_t)(c + 3) * FRAMES));
        }
    } else {
        // asr part: gather columns of t_en (t_en[c*2048 + s]); consecutive s
        // => near-contiguous L2 loads per channel row.
        const int c0 = ch0 - D_DIM;
#pragma unroll 4
        for (int c = 0; c < CH_PER_BLK; ++c) {
            const float* __restrict__ trow = t_en + (size_t)(c0 + c) * SEQ_LEN;
            vfloat4 v;
            v.x = (s4.x >= 0) ? trow[s4.x] : 0.0f;
            v.y = (s4.y >= 0) ? trow[s4.y] : 0.0f;
            v.z = (s4.z >= 0) ? trow[s4.z] : 0.0f;
            v.w = (s4.w >= 0) ? trow[s4.w] : 0.0f;
            __builtin_nontemporal_store(v, (vfloat4*)(outp + (size_t)c * FRAMES));
        }
    }
}

// ---------------------------------------------------------------------------
// Host launch
// ---------------------------------------------------------------------------
extern "C" void kernel_launch(void* const* d_in, const int* in_sizes, int n_in,
                              void* d_out, int out_size, void* d_ws, size_t ws_size,
                              hipStream_t stream)
{
    (void)in_sizes; (void)n_in; (void)out_size; (void)ws_size;

    const float* d        = (const float*)d_in[0];   // [1, 2048, 640]
    const float* t_en     = (const float*)d_in[1];   // [1, 512, 2048]
    const int*   pred_dur = (const int*)d_in[2];     // [2048]
    // d_in[3] = total_frames scalar (fixed 34816; baked into FRAMES)

    int*   map = (int*)d_ws;          // 34816 ints = 139 KB scratch
    float* out = (float*)d_out;       // [1, 1152, 34816]

    build_map_kernel<<<1, 256, 0, stream>>>(pred_dur, map);

    dim3 grid(FRAMES / FR_PER_BLK, N_CH / CH_PER_BLK);   // (34, 36)
    expand_kernel<<<grid, 256, 0, stream>>>(d, t_en, map, out);
}